// DomainProjectionLDP_25194278159054
// MI455X (gfx1250) — compile-verified
//
#include <hip/hip_runtime.h>
#include <hip/hip_bf16.h>

// DomainProjectionLDP for MI455X (gfx1250, wave32, WMMA).
//
// Roofline: 512MB mandatory HBM traffic -> ~22us @ 23.3TB/s. Exact math is
// only 8.6 GFLOP, so we counting-sort rows by domain so every 32-row WMMA
// tile is single-domain (no one-hot 8x FLOP waste) and use the full-f32
// V_WMMA_F32_16X16X4_F32 path (no precision loss vs the f32 reference).
// V and U^T are pre-packed into paired-K float2 layout so every WMMA B
// fragment is a single global_load_b64, and each block processes two M-tiles
// (TM=32) so each B fragment feeds 2 WMMAs.

#define B_  65536
#define D_  1024
#define ND_ 8
#define R_  32
#define TM  32      // rows per block (two 16-row WMMA M-tiles)

typedef float v2f __attribute__((ext_vector_type(2)));
typedef float v8f __attribute__((ext_vector_type(8)));

// D = A(16x4) * B(4x16) + C, full fp32. A: lanes0-15 M=lane {K=k0,k0+1},
// lanes16-31 M=lane-16 {K=k0+2,k0+3}. B mirrored: lane half selects K pair,
// lane&15 selects N. C: vgpr v -> M = v + 8*(lane>>4), N = lane&15.
__device__ __forceinline__ v8f wmma4(v2f a, v2f b, v8f c) {
  return __builtin_amdgcn_wmma_f32_16x16x4_f32(false, a, false, b, (short)0, c,
                                               false, false);
}

// ---- workspace layout (bytes) ----
// [0,256)     : int hdr[64] : [0:8) counts [8:16) offsets [16:24) cursor
//                             [24:33) tile_base (tile_base[8] = total tiles)
// [256,512)   : float regpart[17] (16 orth partials + [16]=nuclear term)
// [512,+256K) : int perm[B]
// +256K,+1M   : v2f Vp[ND][512][32]  : {V[d][2k][r],  V[d][2k+1][r]}
// +1M         : v2f Up[ND][16][1024] : {U[d][c][2k],  U[d][c][2k+1]}
// total ~2.3 MB

__global__ __launch_bounds__(256) void hist_kernel(const int* __restrict__ dom,
                                                   int* hdr) {
  int i = blockIdx.x * 256 + threadIdx.x;
  if (i < B_) atomicAdd(&hdr[dom[i]], 1);
}

__global__ __launch_bounds__(256) void setup_kernel(const float* __restrict__ s,
                                                    int* hdr, float* regpart) {
  __shared__ float red[256];
  int t = threadIdx.x;
  red[t] = fabsf(s[t]);  // ND_*R_ == 256 exactly
  __syncthreads();
  for (int off = 128; off > 0; off >>= 1) {
    if (t < off) red[t] += red[t + off];
    __syncthreads();
  }
  if (t == 0) {
    regpart[16] = 0.1f * red[0] / (float)(R_ * ND_);  // reg_nuc summed /ND
    int off = 0, tb = 0;
    for (int d = 0; d < ND_; ++d) {
      int c = hdr[d];
      hdr[8 + d] = off;    // offsets
      hdr[16 + d] = off;   // cursor (scatter)
      hdr[24 + d] = tb;    // tile base
      off += c;
      tb += (c + TM - 1) / TM;
    }
    hdr[24 + ND_] = tb;    // total tiles (<= 2048+7)
  }
}

__global__ __launch_bounds__(256) void scatter_kernel(const int* __restrict__ dom,
                                                      int* hdr, int* perm) {
  int i = blockIdx.x * 256 + threadIdx.x;
  if (i < B_) {
    int p = atomicAdd(&hdr[16 + dom[i]], 1);
    perm[p] = i;
  }
}

// Pack V and U into paired-K float2 WMMA-B layouts (one b64 per B fragment).
__global__ __launch_bounds__(256) void pack_kernel(const float* __restrict__ V,
                                                   const float* __restrict__ U,
                                                   v2f* __restrict__ Vp,
                                                   v2f* __restrict__ Up) {
  int i = blockIdx.x * 256 + threadIdx.x;  // 0 .. 262143
  if (i < 131072) {
    // Vp[((d*512)+kp)*32 + r] = {V[d][2kp][r], V[d][2kp+1][r]}
    int r = i & 31, kp = (i >> 5) & 511, d = i >> 14;
    const float* src = V + (size_t)d * D_ * R_ + (size_t)(2 * kp) * R_ + r;
    v2f p;
    p.x = src[0];
    p.y = src[R_];
    Vp[i] = p;
  } else {
    // Up[((d*16)+kp)*1024 + col] = {U[d][col][2kp], U[d][col][2kp+1]}
    int j = i - 131072;
    int col = j & 1023, kp = (j >> 10) & 15, d = j >> 14;
    const float* src = U + (size_t)d * D_ * R_ + (size_t)col * R_ + 2 * kp;
    v2f p;
    p.x = src[0];
    p.y = src[1];
    Up[j] = p;
  }
}

// Main fused kernel: one block = one 32-row single-domain tile, 8 waves.
// Pass1: z[32x32] = X[32x1024] @ V_d (K split across waves, 4 WMMA accums)
// LDS reduce + diag(s_d); Pass2: out = feats + z @ U_d^T (N split by wave).
__global__ __launch_bounds__(256) void main_kernel(
    const float* __restrict__ feats, const float* __restrict__ s,
    const v2f* __restrict__ Vp, const v2f* __restrict__ Up,
    const int* __restrict__ hdr, const int* __restrict__ perm,
    float* __restrict__ out) {
  __shared__ int rowIdx[TM];
  __shared__ float zpart[8 * TM * 33];   // per-wave partial z, padded (33KB)
  __shared__ float zred[TM * 33];        // reduced + s-scaled z

  int tile = blockIdx.x;
  if (tile >= hdr[24 + ND_]) return;     // uniform per block
  int d = 0;
#pragma unroll
  for (int i = 1; i < ND_; ++i)
    if (tile >= hdr[24 + i]) d = i;      // zero-count domains collapse safely
  int p0 = hdr[8 + d] + (tile - hdr[24 + d]) * TM;
  int pend = hdr[8 + d] + hdr[d];

  int t = threadIdx.x, wave = t >> 5, lane = t & 31;
  if (t < TM) {
    int p = p0 + t;
    rowIdx[t] = (p < pend) ? perm[p] : -1;
  }
  __syncthreads();

  int mrow = lane & 15;                  // N / M-in-tile position
  int khalf = (lane >> 4) << 1;          // 0 or 2: K sub-position per half-wave
  int ar0 = rowIdx[mrow];       if (ar0 < 0) ar0 = 0;  // garbage never stored
  int ar1 = rowIdx[16 + mrow];  if (ar1 < 0) ar1 = 0;
  const float* ap0 = feats + (size_t)ar0 * D_ + khalf;
  const float* ap1 = feats + (size_t)ar1 * D_ + khalf;
  const v2f* Vpd = Vp + (size_t)d * 512 * 32;

  // ---- pass 1: this wave covers K in [wave*128, wave*128+128) ----
  v8f c00 = {0, 0, 0, 0, 0, 0, 0, 0};    // rows 0-15,  z cols 0-15
  v8f c01 = {0, 0, 0, 0, 0, 0, 0, 0};    // rows 0-15,  z cols 16-31
  v8f c10 = {0, 0, 0, 0, 0, 0, 0, 0};    // rows 16-31, z cols 0-15
  v8f c11 = {0, 0, 0, 0, 0, 0, 0, 0};    // rows 16-31, z cols 16-31
  int kbeg = wave * 128;
#pragma unroll 2
  for (int k0 = kbeg; k0 < kbeg + 128; k0 += 4) {
    v2f a0 = *(const v2f*)(ap0 + k0);    // 8B gather: X[m][k0+khalf .. +1]
    v2f a1 = *(const v2f*)(ap1 + k0);
    int kp = (k0 + khalf) >> 1;          // even K pair index
    v2f b0 = Vpd[kp * 32 + mrow];        // single b64: {V[k][n],V[k+1][n]}
    v2f b1 = Vpd[kp * 32 + 16 + mrow];
    c00 = wmma4(a0, b0, c00);
    c01 = wmma4(a0, b1, c01);
    c10 = wmma4(a1, b0, c10);
    c11 = wmma4(a1, b1, c11);
  }
  {
    float* zp = zpart + wave * (TM * 33);
#pragma unroll
    for (int v = 0; v < 8; ++v) {
      int m = v + ((lane >> 4) << 3);
      zp[m * 33 + mrow] = c00[v];
      zp[m * 33 + 16 + mrow] = c01[v];
      zp[(16 + m) * 33 + mrow] = c10[v];
      zp[(16 + m) * 33 + 16 + mrow] = c11[v];
    }
  }
  __syncthreads();

  // ---- cross-wave reduce + diag(s_d), fixed order (deterministic) ----
  {
    const float* sd = s + d * R_;
#pragma unroll
    for (int e2 = 0; e2 < 4; ++e2) {
      int e = t * 4 + e2;                // 1024 z elements / 256 threads
      int m = e >> 5, r = e & 31;
      float acc = 0.f;
#pragma unroll
      for (int w = 0; w < 8; ++w) acc += zpart[w * (TM * 33) + m * 33 + r];
      zred[m * 33 + r] = acc * sd[r];
    }
  }
  __syncthreads();

  // ---- pass 2: this wave covers output cols [wave*128, wave*128+128) ----
  v2f a2lo[8], a2hi[8];                  // z fragments for both M-tiles (K=32)
#pragma unroll
  for (int j = 0; j < 8; ++j) {
    int kb = 4 * j + khalf;
    a2lo[j].x = zred[mrow * 33 + kb];
    a2lo[j].y = zred[mrow * 33 + kb + 1];
    a2hi[j].x = zred[(16 + mrow) * 33 + kb];
    a2hi[j].y = zred[(16 + mrow) * 33 + kb + 1];
  }
  const v2f* Upd = Up + (size_t)d * 16 * 1024;
  int n0w = wave * 128;
  for (int nt = 0; nt < 8; ++nt) {
    int n0 = n0w + nt * 16;
    v8f clo = {0, 0, 0, 0, 0, 0, 0, 0};
    v8f chi = {0, 0, 0, 0, 0, 0, 0, 0};
#pragma unroll
    for (int j = 0; j < 8; ++j) {
      int kp = (4 * j + khalf) >> 1;
      v2f b = Upd[kp * 1024 + n0 + mrow];  // single b64: {Ut[k][c],Ut[k+1][c]}
      clo = wmma4(a2lo[j], b, clo);
      chi = wmma4(a2hi[j], b, chi);
    }
#pragma unroll
    for (int v = 0; v < 8; ++v) {
      int m = v + ((lane >> 4) << 3);
      int row0 = rowIdx[m];
      if (row0 >= 0) {
        size_t idx = (size_t)row0 * D_ + n0 + mrow;
        out[idx] = feats[idx] + clo[v];  // feats re-read: L2 hit
      }
      int row1 = rowIdx[16 + m];
      if (row1 >= 0) {
        size_t idx = (size_t)row1 * D_ + n0 + mrow;
        out[idx] = feats[idx] + chi[v];
      }
    }
  }
}

// Orthogonality regularizer: block b -> (d = b>>1, matrix = b&1 ? V : U).
// Computes mean((M^T M - I)^2)/ND deterministically into regpart[b].
__global__ __launch_bounds__(256) void reg_kernel(const float* __restrict__ U,
                                                  const float* __restrict__ V,
                                                  float* regpart) {
  __shared__ float chunk[256];
  __shared__ float red[256];
  int b = blockIdx.x;
  const float* Md = ((b & 1) ? V : U) + (size_t)(b >> 1) * D_ * R_;
  int t = threadIdx.x;
  int q = t & 31, rb = t >> 5;           // pairs (rb+8j, q), j=0..3
  float acc0 = 0, acc1 = 0, acc2 = 0, acc3 = 0;
  for (int kc = 0; kc < D_; kc += 8) {
    __syncthreads();
    chunk[t] = Md[(size_t)kc * R_ + t];  // 8 rows of 32, coalesced
    __syncthreads();
#pragma unroll
    for (int kk = 0; kk < 8; ++kk) {
      const float* row = &chunk[kk * 32];
      float xq = row[q];
      acc0 += row[rb] * xq;
      acc1 += row[rb + 8] * xq;
      acc2 += row[rb + 16] * xq;
      acc3 += row[rb + 24] * xq;
    }
  }
  float e0 = acc0 - ((rb == q) ? 1.f : 0.f);
  float e1 = acc1 - ((rb + 8 == q) ? 1.f : 0.f);
  float e2 = acc2 - ((rb + 16 == q) ? 1.f : 0.f);
  float e3 = acc3 - ((rb + 24 == q) ? 1.f : 0.f);
  red[t] = e0 * e0 + e1 * e1 + e2 * e2 + e3 * e3;
  __syncthreads();
  for (int off = 128; off > 0; off >>= 1) {
    if (t < off) red[t] += red[t + off];
    __syncthreads();
  }
  if (t == 0) regpart[b] = red[0] / (float)(R_ * R_ * ND_);
}

__global__ void final_kernel(const float* __restrict__ regpart,
                             float* __restrict__ out) {
  if (threadIdx.x == 0 && blockIdx.x == 0) {
    float acc = regpart[16];             // nuclear term
    for (int i = 0; i < 16; ++i) acc += regpart[i];
    out[(size_t)B_ * D_] = acc;
  }
}

extern "C" void kernel_launch(void* const* d_in, const int* in_sizes, int n_in,
                              void* d_out, int out_size, void* d_ws,
                              size_t ws_size, hipStream_t stream) {
  (void)in_sizes; (void)n_in; (void)out_size; (void)ws_size;
  const float* feats = (const float*)d_in[0];
  const int* dom = (const int*)d_in[1];
  const float* U = (const float*)d_in[2];
  const float* s = (const float*)d_in[3];
  const float* V = (const float*)d_in[4];
  float* out = (float*)d_out;

  char* ws = (char*)d_ws;                        // needs ~2.3 MB
  int* hdr = (int*)ws;
  float* regpart = (float*)(ws + 256);
  int* perm = (int*)(ws + 512);
  v2f* Vp = (v2f*)(ws + 512 + (size_t)B_ * sizeof(int));
  v2f* Up = (v2f*)(ws + 512 + (size_t)B_ * sizeof(int) +
                   (size_t)ND_ * 512 * 32 * sizeof(v2f));

  hipMemsetAsync(hdr, 0, 256, stream);
  hist_kernel<<<B_ / 256, 256, 0, stream>>>(dom, hdr);
  setup_kernel<<<1, 256, 0, stream>>>(s, hdr, regpart);
  scatter_kernel<<<B_ / 256, 256, 0, stream>>>(dom, hdr, perm);
  pack_kernel<<<1024, 256, 0, stream>>>(V, U, Vp, Up);
  main_kernel<<<(B_ / TM) + ND_, 256, 0, stream>>>(feats, s, Vp, Up, hdr, perm,
                                                   out);
  reg_kernel<<<2 * ND_, 256, 0, stream>>>(U, V, regpart);
  final_kernel<<<1, 32, 0, stream>>>(regpart, out);
}